// LSM_45260365365524
// MI455X (gfx1250) — compile-verified
//
#include <hip/hip_runtime.h>
#include <hip/hip_bf16.h>

typedef __attribute__((ext_vector_type(2))) float v2f;
typedef __attribute__((ext_vector_type(8))) float v8f;

// ---------------- wave32 reductions ----------------
__device__ __forceinline__ float wred_sum(float v) {
    v += __shfl_xor(v, 16, 32);
    v += __shfl_xor(v, 8, 32);
    v += __shfl_xor(v, 4, 32);
    v += __shfl_xor(v, 2, 32);
    v += __shfl_xor(v, 1, 32);
    return v;
}
__device__ __forceinline__ float wred_max(float v) {
    v = fmaxf(v, __shfl_xor(v, 16, 32));
    v = fmaxf(v, __shfl_xor(v, 8, 32));
    v = fmaxf(v, __shfl_xor(v, 4, 32));
    v = fmaxf(v, __shfl_xor(v, 2, 32));
    v = fmaxf(v, __shfl_xor(v, 1, 32));
    return v;
}

// fast (~1 ulp) hardware transcendentals: v_sqrt_f32 / v_rcp_f32
__device__ __forceinline__ float fsqrt(float x) { return __builtin_amdgcn_sqrtf(x); }
__device__ __forceinline__ float frcp(float x)  { return __builtin_amdgcn_rcpf(x); }

// In-register softmax of one 64-wide row: lane holds elements [2*lane, 2*lane+1].
__device__ __forceinline__ v2f softmax_row(const float* rowBase, int lane) {
    v2f v = *(const v2f*)(rowBase + lane * 2);
    float m = wred_max(fmaxf(v.x, v.y));
    float e0 = __expf(v.x - m);
    float e1 = __expf(v.y - m);
    float inv = frcp(wred_sum(e0 + e1));
    v2f r; r.x = e0 * inv; r.y = e1 * inv;
    return r;
}

// ---------------- kernel 1: gather + softmax sampled rows ----------------
// one wave per sampled row; S must be a multiple of 8 (4096 is)
__global__ void lsm_prep(const float* __restrict__ z1,
                         const float* __restrict__ gamma,
                         const int*   __restrict__ sample_idx,
                         float* __restrict__ zs,
                         float* __restrict__ sq,
                         float* __restrict__ g) {
    const int lane = threadIdx.x & 31;
    const int w = blockIdx.x * 8 + (threadIdx.x >> 5);   // row in [0,S)
    const int idx = sample_idx[w];
    v2f z = softmax_row(z1 + (size_t)idx * 64, lane);
    *(v2f*)(zs + (size_t)w * 64 + lane * 2) = z;
    float q = wred_sum(z.x * z.x + z.y * z.y);
    if (lane == 0) {
        sq[w] = q;
        g[w] = __expf(gamma[idx]);
    }
}

// ---------------- kernel 2: WMMA pairwise-block term ----------------
// wave computes a 32x32 patch of zs @ zs^T via V_WMMA_F32_16X16X4_F32,
// then fuses distance/exp/g-weighting and reduces to one partial per block.
__device__ __forceinline__ float tile_post(const v8f& c, int mbase, int nbase,
                                           int half, int lr, float rl,
                                           const float* __restrict__ sq,
                                           const float* __restrict__ g) {
    float s = 0.f;
    const int n = nbase + lr;
    const float sqn = sq[n];
    const float gn = g[n];
#pragma unroll
    for (int r = 0; r < 8; ++r) {
        int m = mbase + r + 8 * half;
        if (m != n) {
            float d2 = sq[m] + sqn - 2.0f * c[r];
            d2 = fmaxf(d2, 0.0f);
            s += g[m] * gn * __expf(-rl * fsqrt(d2));
        }
    }
    return s;
}

__global__ void __launch_bounds__(256, 1)
lsm_pdist1(const float* __restrict__ zs,
           const float* __restrict__ sq,
           const float* __restrict__ g,
           const float* __restrict__ reg_l,
           float* __restrict__ partials,
           int t32) {
    const int lane = threadIdx.x & 31;
    const int wid = threadIdx.x >> 5;
    const int w = blockIdx.x * 8 + wid;       // wave-tile id, grid is exact
    const int wm = w / t32;
    const int wn = w % t32;
    const int half = lane >> 4;               // 0 or 1
    const int lr = lane & 15;
    const float rl = reg_l[0];

    const float* pr0 = zs + (size_t)(wm * 32 + lr) * 64 + 2 * half;  // A rows (M)
    const float* pr1 = pr0 + 16 * 64;
    const float* pc0 = zs + (size_t)(wn * 32 + lr) * 64 + 2 * half;  // B cols (N)
    const float* pc1 = pc0 + 16 * 64;

    v8f acc00 = {}; v8f acc01 = {}; v8f acc10 = {}; v8f acc11 = {};

    // unroll-limited: keeps in-flight loads at ~2 iterations worth, no spills
#pragma unroll 2
    for (int kb = 0; kb < 16; ++kb) {
        const int off = kb * 4;               // this lane's K pair base
        v2f a0 = *(const v2f*)(pr0 + off);
        v2f a1 = *(const v2f*)(pr1 + off);
        v2f b0 = *(const v2f*)(pc0 + off);
        v2f b1 = *(const v2f*)(pc1 + off);
        acc00 = __builtin_amdgcn_wmma_f32_16x16x4_f32(false, a0, false, b0, (short)0, acc00, false, false);
        acc01 = __builtin_amdgcn_wmma_f32_16x16x4_f32(false, a0, false, b1, (short)0, acc01, false, false);
        acc10 = __builtin_amdgcn_wmma_f32_16x16x4_f32(false, a1, false, b0, (short)0, acc10, false, false);
        acc11 = __builtin_amdgcn_wmma_f32_16x16x4_f32(false, a1, false, b1, (short)0, acc11, false, false);
    }

    float lsum = 0.f;
    lsum += tile_post(acc00, wm * 32,      wn * 32,      half, lr, rl, sq, g);
    lsum += tile_post(acc01, wm * 32,      wn * 32 + 16, half, lr, rl, sq, g);
    lsum += tile_post(acc10, wm * 32 + 16, wn * 32,      half, lr, rl, sq, g);
    lsum += tile_post(acc11, wm * 32 + 16, wn * 32 + 16, half, lr, rl, sq, g);

    __shared__ float sm[8];
    float wsum = wred_sum(lsum);
    if (lane == 0) sm[wid] = wsum;
    __syncthreads();
    if (threadIdx.x == 0) {
        float t = 0.f;
#pragma unroll
        for (int i = 0; i < 8; ++i) t += sm[i];
        partials[blockIdx.x] = -0.5f * t;     // enters final sum with -1/2 weight
    }
}

// ---------------- kernel 3: observed-edge term ----------------
// one wave per edge, 16 edges per wave, softmax recomputed on the fly.
#define EPW 16
__global__ void lsm_edges(const float* __restrict__ z1,
                          const float* __restrict__ gamma,
                          const float* __restrict__ reg_l,
                          const int* __restrict__ si,
                          const int* __restrict__ sj,
                          float* __restrict__ partials,
                          int E) {
    const int lane = threadIdx.x & 31;
    const int wid = threadIdx.x >> 5;
    const float rl = reg_l[0];
    const long base = ((long)blockIdx.x * 8 + wid) * EPW;

    float acc = 0.f;   // meaningful on lane 0 only
    for (int t = 0; t < EPW; ++t) {
        long e = base + t;
        if (e < E) {                          // uniform across the wave
            int i = si[e];
            int j = sj[e];
            v2f zi = softmax_row(z1 + (size_t)i * 64, lane);
            v2f zj = softmax_row(z1 + (size_t)j * 64, lane);
            float dx = zi.x - zj.x + 1e-6f;
            float dy = zi.y - zj.y + 1e-6f;
            float s2 = wred_sum(dx * dx + dy * dy);
            if (lane == 0)
                acc += -rl * fsqrt(s2) + gamma[i] + gamma[j];
        }
    }

    __shared__ float sm[8];
    float wsum = wred_sum(acc);
    if (lane == 0) sm[wid] = wsum;
    __syncthreads();
    if (threadIdx.x == 0) {
        float t = 0.f;
#pragma unroll
        for (int i = 0; i < 8; ++i) t += sm[i];
        partials[blockIdx.x] = t;
    }
}

// ---------------- kernel 4: deterministic final reduction ----------------
__global__ void lsm_final(const float* __restrict__ partials, int n, float* __restrict__ out) {
    __shared__ float sm[256];
    float s = 0.f;
    for (int i = threadIdx.x; i < n; i += 256) s += partials[i];
    sm[threadIdx.x] = s;
    __syncthreads();
    for (int st = 128; st > 0; st >>= 1) {
        if (threadIdx.x < st) sm[threadIdx.x] += sm[threadIdx.x + st];
        __syncthreads();
    }
    if (threadIdx.x == 0) out[0] = sm[0];
}

extern "C" void kernel_launch(void* const* d_in, const int* in_sizes, int n_in,
                              void* d_out, int out_size, void* d_ws, size_t ws_size,
                              hipStream_t stream) {
    const float* latent_z1  = (const float*)d_in[0];
    const float* gamma      = (const float*)d_in[1];
    const float* reg_l      = (const float*)d_in[2];
    const int*   sample_idx = (const int*)d_in[3];
    const int*   si         = (const int*)d_in[4];
    const int*   sj         = (const int*)d_in[5];

    const int S = in_sizes[3];          // 4096
    const int E = in_sizes[4];          // 1,000,000

    float* ws = (float*)d_ws;
    float* zs = ws;                          // S*64 floats
    float* sq = zs + (size_t)S * 64;         // S floats
    float* g  = sq + S;                      // S floats
    float* partials = g + S;

    const int t32 = S / 32;                  // 128
    const int gemmBlocks = (t32 * t32) / 8;  // 2048 (exact)
    const int edgeBlocks = (E + (8 * EPW) - 1) / (8 * EPW);
    const int np = gemmBlocks + edgeBlocks;

    lsm_prep  <<<S / 8,      256, 0, stream>>>(latent_z1, gamma, sample_idx, zs, sq, g);
    lsm_pdist1<<<gemmBlocks, 256, 0, stream>>>(zs, sq, g, reg_l, partials, t32);
    lsm_edges <<<edgeBlocks, 256, 0, stream>>>(latent_z1, gamma, reg_l, si, sj,
                                               partials + gemmBlocks, E);
    lsm_final <<<1,          256, 0, stream>>>(partials, np, (float*)d_out);
}